// HyperTurbulentGenerator_76897094467970
// MI455X (gfx1250) — compile-verified
//
#include <hip/hip_runtime.h>
#include <hip/hip_bf16.h>

// HyperTurbulentGenerator for MI455X (gfx1250), wave32, WMMA f32 16x16x4.
// B=32 batches -> 32 workgroups. D=64, N=8, S=128.
// Key algebraic reduction: z[b,s] = M^(s+1) @ z0[b], so we never build the
// (B,S,64,64) propagator tensor; we build M = expm(A dt + eps I) via a
// Horner-form Taylor series (||A dt|| ~ 5e-3 -> 12 terms is ample), then
// M^16 via 4 squarings, then advance 16-wide vector blocks with WMMA.

typedef __attribute__((ext_vector_type(2))) float v2f;
typedef __attribute__((ext_vector_type(8))) float v8f;

#define MS 65   // 64x64 matrix LDS row stride (odd -> conflict-free columns)
#define VS 17   // 64x16 vector-block LDS row stride
#define NTERMS 12
#define DD 64
#define SS 128

// C(64x64) = op(X) @ Y, all LDS, full 8-wave workgroup, 16 output tiles
// (2 per wave), K accumulated in 16 chunks of 4 via V_WMMA_F32_16X16X4_F32.
template <bool TX>
__device__ __forceinline__ void mm64(const float* X, const float* Y, float* Z,
                                     int wave, int lane)
{
    __syncthreads();
    const int m  = lane & 15;   // M (A) / N (B,C) index within tile
    const int hi = lane >> 4;   // selects upper half of K (A,B) / M (C)
    #pragma unroll
    for (int rep = 0; rep < 2; ++rep) {
        const int t  = wave + rep * 8;          // tile 0..15
        const int ti = t >> 2, tj = t & 3;
        const int row = ti * 16 + m;
        const int col = tj * 16 + m;
        v8f acc = {};
        #pragma unroll
        for (int kc = 0; kc < 16; ++kc) {
            const int k0 = kc * 4 + hi * 2;
            v2f a, b;
            if (TX) {   // A[m][k] = X[k][m]  (transposed operand)
                a.x = X[(k0 + 0) * MS + row];
                a.y = X[(k0 + 1) * MS + row];
            } else {
                a.x = X[row * MS + k0 + 0];
                a.y = X[row * MS + k0 + 1];
            }
            b.x = Y[(k0 + 0) * MS + col];
            b.y = Y[(k0 + 1) * MS + col];
            acc = __builtin_amdgcn_wmma_f32_16x16x4_f32(false, a, false, b,
                                                        (short)0, acc,
                                                        false, false);
        }
        #pragma unroll
        for (int r = 0; r < 8; ++r)
            Z[(ti * 16 + r + 8 * hi) * MS + tj * 16 + m] = acc[r];
    }
    __syncthreads();
}

// W(64x16) = X(64x64) @ V(64x16), waves 0..3 each own one 16x16 output tile.
__device__ __forceinline__ void mm64x16(const float* X, const float* V, float* W,
                                        int wave, int lane)
{
    __syncthreads();
    if (wave < 4) {             // wave-uniform branch: EXEC stays all-ones
        const int m  = lane & 15;
        const int hi = lane >> 4;
        const int row = wave * 16 + m;
        v8f acc = {};
        #pragma unroll
        for (int kc = 0; kc < 16; ++kc) {
            const int k0 = kc * 4 + hi * 2;
            v2f a, b;
            a.x = X[row * MS + k0 + 0];
            a.y = X[row * MS + k0 + 1];
            b.x = V[(k0 + 0) * VS + m];
            b.y = V[(k0 + 1) * VS + m];
            acc = __builtin_amdgcn_wmma_f32_16x16x4_f32(false, a, false, b,
                                                        (short)0, acc,
                                                        false, false);
        }
        #pragma unroll
        for (int r = 0; r < 8; ++r)
            W[(wave * 16 + r + 8 * hi) * VS + m] = acc[r];
    }
    __syncthreads();
}

__global__ __launch_bounds__(256)
void htg_kernel(const float* __restrict__ z0,
                const float* __restrict__ ts,
                const float* __restrict__ kcf,
                const float* __restrict__ rcf,
                const float* __restrict__ alpha,
                const float* __restrict__ beta,
                const float* __restrict__ Kb,
                const float* __restrict__ Rb,
                float2* __restrict__ out)
{
    __shared__ float B0[DD * MS];
    __shared__ float B1[DD * MS];
    __shared__ float B2[DD * MS];
    __shared__ float Vb[DD * VS];
    __shared__ float Vn[DD * VS];
    __shared__ float vec[2][DD];

    const int b    = blockIdx.x;
    const int tid  = threadIdx.x;
    const int wave = tid >> 5;
    const int lane = tid & 31;

    const float dt = ts[0];
    const float al = alpha[b];
    const float be = beta[b];

    float kco[8], rco[8];
    #pragma unroll
    for (int n = 0; n < 8; ++n) {
        kco[n] = kcf[b * 8 + n];
        rco[n] = rcf[b * 8 + n];
    }

    // ---- weighted basis sums: B0 = K_sum, B1 = R_sum -------------------
    for (int idx = tid; idx < DD * DD; idx += 256) {
        const int i = idx >> 6, j = idx & 63;
        float ks = 0.f, rs = 0.f;
        #pragma unroll
        for (int n = 0; n < 8; ++n) {
            ks = fmaf(kco[n], Kb[n * 4096 + idx], ks);
            rs = fmaf(rco[n], Rb[n * 4096 + idx], rs);
        }
        B0[i * MS + j] = ks;
        B1[i * MS + j] = rs;
    }

    // ---- B2 = R^T @ R (A operand transposed) ---------------------------
    mm64<true>(B1, B1, B2, wave, lane);

    // ---- X = dt*(al*(K - K^T) - be*(R^T R)) + eps*I -> B1 --------------
    for (int idx = tid; idx < DD * DD; idx += 256) {
        const int i = idx >> 6, j = idx & 63;
        float x = al * (B0[i * MS + j] - B0[j * MS + i]) - be * B2[i * MS + j];
        x = x * dt + ((i == j) ? 1e-6f : 0.f);
        B1[i * MS + j] = x;
    }
    __syncthreads();   // B0 fully read before it is reused as T

    // ---- Horner Taylor: T = I; for k=N..1: T = I + (X@T)/k -------------
    // init T = I + X/NTERMS (first Horner step fused, saves one matmul)
    for (int idx = tid; idx < DD * DD; idx += 256) {
        const int i = idx >> 6, j = idx & 63;
        B0[idx ? (i * MS + j) : 0] =
            ((i == j) ? 1.f : 0.f) + B1[i * MS + j] * (1.f / (float)NTERMS);
    }
    for (int k = NTERMS - 1; k >= 1; --k) {
        mm64<false>(B1, B0, B2, wave, lane);     // B2 = X @ T
        const float inv = 1.f / (float)k;
        for (int idx = tid; idx < DD * DD; idx += 256) {
            const int i = idx >> 6, j = idx & 63;
            B0[i * MS + j] = ((i == j) ? 1.f : 0.f) + B2[i * MS + j] * inv;
        }
    }
    // M = expm(X) in B0

    // ---- M^16 via 4 squarings -> B2 ------------------------------------
    mm64<false>(B0, B0, B1, wave, lane);   // M^2
    mm64<false>(B1, B1, B2, wave, lane);   // M^4
    mm64<false>(B2, B2, B1, wave, lane);   // M^8
    mm64<false>(B1, B1, B2, wave, lane);   // M^16

    // ---- first 16 steps: sequential matvec v_{p} = M v_{p-1} -----------
    if (tid < DD) vec[0][tid] = z0[b * DD + tid];
    __syncthreads();
    int cur = 0;
    for (int p = 1; p <= 16; ++p) {
        if (tid < DD) {
            float acc = 0.f;
            #pragma unroll 8
            for (int j = 0; j < DD; ++j)
                acc = fmaf(B0[tid * MS + j], vec[cur][j], acc);
            vec[cur ^ 1][tid] = acc;
            Vb[tid * VS + (p - 1)] = acc;                 // column p-1
            out[((size_t)b * SS + (p - 1)) * DD + tid] = make_float2(acc, 0.f);
        }
        __syncthreads();
        cur ^= 1;
    }

    // ---- remaining 7 chunks of 16 steps: V <- M^16 @ V (WMMA) ----------
    float* V = Vb;
    float* W = Vn;
    for (int c = 1; c < 8; ++c) {
        mm64x16(B2, V, W, wave, lane);
        for (int idx = tid; idx < DD * 16; idx += 256) {
            const int i = idx >> 4, j = idx & 15;
            out[((size_t)b * SS + (16 * c + j)) * DD + i] =
                make_float2(W[i * VS + j], 0.f);
        }
        float* tmp = V; V = W; W = tmp;
    }
}

extern "C" void kernel_launch(void* const* d_in, const int* in_sizes, int n_in,
                              void* d_out, int out_size, void* d_ws, size_t ws_size,
                              hipStream_t stream) {
    (void)in_sizes; (void)n_in; (void)d_ws; (void)ws_size; (void)out_size;
    const float* z0    = (const float*)d_in[0];
    const float* ts    = (const float*)d_in[1];
    const float* kcf   = (const float*)d_in[2];
    const float* rcf   = (const float*)d_in[3];
    const float* alpha = (const float*)d_in[4];
    const float* beta  = (const float*)d_in[5];
    const float* Kb    = (const float*)d_in[6];
    const float* Rb    = (const float*)d_in[7];
    float2* out = (float2*)d_out;
    htg_kernel<<<32, 256, 0, stream>>>(z0, ts, kcf, rcf, alpha, beta, Kb, Rb, out);
}